// Sampler_30245159698948
// MI455X (gfx1250) — compile-verified
//
#include <hip/hip_runtime.h>
#include <stdint.h>

#define TPB 256
#define NBUF 3

// ---- CDNA5 async global->LDS path (guarded; fallback = sync copy) ----
#if defined(__AMDGCN__) && __has_builtin(__builtin_amdgcn_global_load_async_to_lds_b128)
#define HAVE_ASYNC_LDS 1
#else
#define HAVE_ASYNC_LDS 0
#endif

#if HAVE_ASYNC_LDS
typedef int v4i_b128 __attribute__((vector_size(16)));
typedef v4i_b128 __attribute__((address_space(1))) gbl_v4i;  // global
typedef v4i_b128 __attribute__((address_space(3))) lds_v4i;  // LDS
#endif

__device__ __forceinline__ void stage_load16(const float4* g, float4* l) {
#if HAVE_ASYNC_LDS
    __builtin_amdgcn_global_load_async_to_lds_b128(
        (gbl_v4i*)g, (lds_v4i*)l, /*offset=*/0, /*cpol=*/0);
#else
    *l = *g;
#endif
}

template <int N>
__device__ __forceinline__ void wait_asynccnt_le() {
#if HAVE_ASYNC_LDS
#if __has_builtin(__builtin_amdgcn_s_wait_asynccnt)
    __builtin_amdgcn_s_wait_asynccnt((short)N);
#else
    asm volatile("s_wait_asynccnt %0" ::"n"(N) : "memory");
#endif
    asm volatile("" ::: "memory");  // keep LDS reads after the wait
#endif
}

// ---- JAX threefry2x32 with key = jax.random.key(42) -> (k0=0, k1=42) ----
__device__ __forceinline__ void threefry2x32_0_42(uint32_t c0, uint32_t c1,
                                                  uint32_t& o0, uint32_t& o1) {
    const uint32_t k0 = 0u, k1 = 42u;
    const uint32_t k2 = k0 ^ k1 ^ 0x1BD11BDAu;
    uint32_t x0 = c0 + k0, x1 = c1 + k1;
#define TF_RND(r)                          \
    x0 += x1;                              \
    x1 = (x1 << (r)) | (x1 >> (32 - (r))); \
    x1 ^= x0;
    TF_RND(13) TF_RND(15) TF_RND(26) TF_RND(6)
    x0 += k1; x1 += k2 + 1u;
    TF_RND(17) TF_RND(29) TF_RND(16) TF_RND(24)
    x0 += k2; x1 += k0 + 2u;
    TF_RND(13) TF_RND(15) TF_RND(26) TF_RND(6)
    x0 += k0; x1 += k1 + 3u;
    TF_RND(17) TF_RND(29) TF_RND(16) TF_RND(24)
    x0 += k1; x1 += k2 + 4u;
    TF_RND(13) TF_RND(15) TF_RND(26) TF_RND(6)
    x0 += k2; x1 += k0 + 5u;
#undef TF_RND
    o0 = x0;
    o1 = x1;
}

// JAX: u = bitcast((bits>>9)|0x3F800000) - 1; uniform(minval=tiny): max(tiny, u+tiny)
__device__ __forceinline__ float gumbel_from_bits(uint32_t bits) {
    float u = __uint_as_float((bits >> 9) | 0x3F800000u) - 1.0f;
    u = fmaxf(1.17549435e-38f, u + 1.17549435e-38f);
    return -logf(-logf(u));
}

// scalar accumulator update (all refs are SROA-able locals after inlining)
__device__ __forceinline__ void upd(float raw, float scaled, float g, int v,
                                    float& gval, int& gidx,
                                    float& sval, int& sidx, float& ssc,
                                    float& m, float& s) {
    if (raw > gval) { gval = raw; gidx = v; }
    const float sv = scaled + g;
    if (sv > sval) { sval = sv; sidx = v; ssc = scaled; }
    if (scaled > m) {
        s = s * __expf(m - scaled) + 1.0f;
        m = scaled;
    } else {
        s += __expf(scaled - m);
    }
}

__global__ __launch_bounds__(TPB) void Sampler_30245159698948_kernel(
    const float* __restrict__ logits, const float* __restrict__ temps,
    float* __restrict__ out, int B, int V) {
    const int tid = threadIdx.x;
    const int row0 = blockIdx.x;
    const int rows_half = gridDim.x;  // B/2
    const int row1 = row0 + rows_half;
    const uint32_t HALF = (uint32_t)rows_half * (uint32_t)V;  // threefry counter split

    const float t0 = temps[row0];
    const float t1 = temps[row1];
    const float st0 = (t0 == 0.0f) ? 1.0f : t0;
    const float st1 = (t1 == 0.0f) ? 1.0f : t1;

    const float4* g0 = (const float4*)(logits + (size_t)row0 * (size_t)V);
    const float4* g1 = (const float4*)(logits + (size_t)row1 * (size_t)V);

    __shared__ float4 stage[NBUF][2][TPB];  // 24 KB of 320 KB LDS
    __shared__ float sh_gval[TPB];
    __shared__ int   sh_gidx[TPB];
    __shared__ float sh_sval[TPB];
    __shared__ int   sh_sidx[TPB];
    __shared__ float sh_ssc[TPB];
    __shared__ float sh_m[TPB];
    __shared__ float sh_s[TPB];

    const float NEGINF = -__builtin_inff();
    float gval0 = NEGINF, sval0 = NEGINF, ssc0 = 0.0f, m0 = NEGINF, s0 = 0.0f;
    float gval1 = NEGINF, sval1 = NEGINF, ssc1 = 0.0f, m1 = NEGINF, s1 = 0.0f;
    int gidx0 = 0, sidx0 = 0, gidx1 = 0, sidx1 = 0;

    const int iters = V / (4 * TPB);  // 125 for V=128000

    // prologue: stage tiles 0 and 1 (prefetch distance 2)
    stage_load16(g0 + tid, &stage[0][0][tid]);
    stage_load16(g1 + tid, &stage[0][1][tid]);
    if (1 < iters) {
        stage_load16(g0 + (TPB + tid), &stage[1][0][tid]);
        stage_load16(g1 + (TPB + tid), &stage[1][1][tid]);
    }

    int buf = 0;
    for (int it = 0; it < iters; ++it) {
        if (it + 2 < iters) {
            const int j2 = (it + 2) * TPB + tid;
            int nb = buf + 2; if (nb >= NBUF) nb -= NBUF;
            stage_load16(g0 + j2, &stage[nb][0][tid]);
            stage_load16(g1 + j2, &stage[nb][1][tid]);
            wait_asynccnt_le<4>();  // tile `it` done; it+1, it+2 in flight
        } else if (it + 1 < iters) {
            wait_asynccnt_le<2>();
        } else {
            wait_asynccnt_le<0>();
        }

        const float4 r0 = stage[buf][0][tid];
        const float4 r1 = stage[buf][1][tid];
        const int j = it * TPB + tid;   // float4 index within row
        const int v0 = j * 4;           // column
        const uint32_t base = (uint32_t)row0 * (uint32_t)V + (uint32_t)v0;

        float eA[4] = {r0.x, r0.y, r0.z, r0.w};
        float eB[4] = {r1.x, r1.y, r1.z, r1.w};
#pragma unroll
        for (int e = 0; e < 4; ++e) {
            uint32_t o0, o1;
            // one threefry covers element (row0,v) and its pair (row0+B/2,v)
            threefry2x32_0_42(base + (uint32_t)e, base + (uint32_t)e + HALF, o0, o1);
            const float gA = gumbel_from_bits(o0);
            const float gB = gumbel_from_bits(o1);
            const float xA = eA[e];
            const float xB = eB[e];
            upd(xA, xA / st0, gA, v0 + e, gval0, gidx0, sval0, sidx0, ssc0, m0, s0);
            upd(xB, xB / st1, gB, v0 + e, gval1, gidx1, sval1, sidx1, ssc1, m1, s1);
        }
        ++buf; if (buf == NBUF) buf = 0;
    }

    // block reduction: row0 then row1 (reuse shared arrays)
    for (int r = 0; r < 2; ++r) {
        sh_gval[tid] = (r == 0) ? gval0 : gval1;
        sh_gidx[tid] = (r == 0) ? gidx0 : gidx1;
        sh_sval[tid] = (r == 0) ? sval0 : sval1;
        sh_sidx[tid] = (r == 0) ? sidx0 : sidx1;
        sh_ssc[tid]  = (r == 0) ? ssc0  : ssc1;
        sh_m[tid]    = (r == 0) ? m0    : m1;
        sh_s[tid]    = (r == 0) ? s0    : s1;
        __syncthreads();
        for (int off = TPB / 2; off > 0; off >>= 1) {
            if (tid < off) {
                const float bg = sh_gval[tid + off]; const int bgi = sh_gidx[tid + off];
                if (bg > sh_gval[tid] || (bg == sh_gval[tid] && bgi < sh_gidx[tid])) {
                    sh_gval[tid] = bg; sh_gidx[tid] = bgi;
                }
                const float bs = sh_sval[tid + off]; const int bsi = sh_sidx[tid + off];
                if (bs > sh_sval[tid] || (bs == sh_sval[tid] && bsi < sh_sidx[tid])) {
                    sh_sval[tid] = bs; sh_sidx[tid] = bsi; sh_ssc[tid] = sh_ssc[tid + off];
                }
                const float am = sh_m[tid], bm = sh_m[tid + off];
                const float mm = fmaxf(am, bm);
                sh_s[tid] = sh_s[tid] * __expf(am - mm) + sh_s[tid + off] * __expf(bm - mm);
                sh_m[tid] = mm;
            }
            __syncthreads();
        }
        if (tid == 0) {
            const int row = (r == 0) ? row0 : row1;
            const float t = temps[row];
            const int token = (t == 0.0f) ? sh_gidx[0] : sh_sidx[0];
            // if t==0, safe_t==1 so scaled==raw, and sh_gval[0] is the scaled max
            const float sel = (t == 0.0f) ? sh_gval[0] : sh_ssc[0];
            const float lp = sel - (sh_m[0] + logf(sh_s[0]));
            out[row] = (float)token;       // tokens (output 0), stored as f32
            out[B + row] = lp;             // log_prob_tokens (output 1)
        }
        __syncthreads();
    }
}

extern "C" void kernel_launch(void* const* d_in, const int* in_sizes, int n_in,
                              void* d_out, int out_size, void* d_ws, size_t ws_size,
                              hipStream_t stream) {
    (void)n_in; (void)out_size; (void)d_ws; (void)ws_size;
    const float* logits = (const float*)d_in[0];
    const float* temps = (const float*)d_in[1];
    float* out = (float*)d_out;
    const int B = in_sizes[1];               // 1024
    const int V = in_sizes[0] / B;           // 128000
    dim3 grid(B / 2), block(TPB);
    Sampler_30245159698948_kernel<<<grid, block, 0, stream>>>(logits, temps, out, B, V);
}